// ConjugatedGrad_18305150616038
// MI455X (gfx1250) — compile-verified
//
#include <hip/hip_runtime.h>
#include <math.h>

#define N   320
#define NN  (N*N)
#define BB  8
#define CC  16
#define NITER 10
#define EPSV 1e-10f

typedef float v2f __attribute__((ext_vector_type(2)));
typedef float v8f __attribute__((ext_vector_type(8)));

__device__ __forceinline__ v8f wmma_f32(v2f a, v2f b, v8f c) {
  // V_WMMA_F32_16X16X4_F32 : D = A(16x4) * B(4x16) + C(16x16)
  return __builtin_amdgcn_wmma_f32_16x16x4_f32(false, a, false, b, (short)0, c, false, false);
}

// ---------------------------------------------------------------------------
// Batched complex GEMM  C[b] (op) = A[b] * B[b], all 320x320 row-major.
// CONJA/CONJB: conjugate operand (folded into LDS tile load).
// EPI: 0 = store, 1 = store * mask[b], 2 = Ap += conj(csm[b,coil]) .* result
// Block: 128 threads (4 waves), 64x64 output tile, K-tile 16.
// ---------------------------------------------------------------------------
template<int CONJA, int CONJB, int EPI>
__global__ void __launch_bounds__(128)
zgemm_wmma(const float* __restrict__ Agr, const float* __restrict__ Agi, unsigned sA,
           const float* __restrict__ Bgr, const float* __restrict__ Bgi, unsigned sB,
           float* __restrict__ Cre, float* __restrict__ Cim,
           const int* __restrict__ mask,
           const float* __restrict__ csm, int coil,
           const float* __restrict__ done)
{
  if (done[0] != 0.0f) return;   // uniform: EXEC stays all-ones for WMMA

  __shared__ float Als[2][64][17];   // [re/im][m][k], pad 17 kills bank conflicts
  __shared__ float Bls[2][16][64];   // [re/im][k][n]

  const int tid  = threadIdx.x;
  const int wave = tid >> 5;
  const int lane = tid & 31;
  const int half = lane >> 4;    // K-pair select (ISA f32 A/B layout)
  const int ln   = lane & 15;
  const int b    = blockIdx.z;
  const int m0   = blockIdx.y * 64;
  const int n0   = blockIdx.x * 64;

  const float* Ar = Agr + (size_t)b * sA;
  const float* Ai = Agi + (size_t)b * sA;
  const float* Br = Bgr + (size_t)b * sB;
  const float* Bi = Bgi + (size_t)b * sB;

  v8f accr[4], acci[4];
  const v8f vz = {0.f,0.f,0.f,0.f,0.f,0.f,0.f,0.f};
#pragma unroll
  for (int t = 0; t < 4; ++t) { accr[t] = vz; acci[t] = vz; }

  for (int k0 = 0; k0 < N; k0 += 16) {
    __syncthreads();
    if (k0 + 16 < N) {  // global_prefetch_b8 of next K tiles
      __builtin_prefetch(&Ar[(size_t)(m0 + (tid >> 1)) * N + k0 + 16], 0, 1);
      __builtin_prefetch(&Br[(size_t)(k0 + 16 + (tid & 15)) * N + n0], 0, 1);
    }
    for (int i = tid; i < 1024; i += 128) {       // A tile 64x16 complex
      int r = i >> 4, c = i & 15;
      float re = Ar[(size_t)(m0 + r) * N + (k0 + c)];
      float im = Ai[(size_t)(m0 + r) * N + (k0 + c)];
      Als[0][r][c] = re;
      Als[1][r][c] = CONJA ? -im : im;
    }
    for (int i = tid; i < 1024; i += 128) {       // B tile 16x64 complex
      int r = i >> 6, c = i & 63;
      float re = Br[(size_t)(k0 + r) * N + (n0 + c)];
      float im = Bi[(size_t)(k0 + r) * N + (n0 + c)];
      Bls[0][r][c] = re;
      Bls[1][r][c] = CONJB ? -im : im;
    }
    __syncthreads();

#pragma unroll
    for (int k4 = 0; k4 < 4; ++k4) {
      const int kk   = k4 * 4 + half * 2;
      const int arow = wave * 16 + ln;
      v2f ar = { Als[0][arow][kk], Als[0][arow][kk + 1] };
      v2f ai = { Als[1][arow][kk], Als[1][arow][kk + 1] };
#pragma unroll
      for (int nt = 0; nt < 4; ++nt) {
        const int cl = nt * 16 + ln;
        v2f br  = { Bls[0][kk][cl], Bls[0][kk + 1][cl] };
        v2f bi  = { Bls[1][kk][cl], Bls[1][kk + 1][cl] };
        v2f nbi = -bi;  // F32 WMMA has no A/B NEG modifier -> negate fragment
        accr[nt] = wmma_f32(ar, br,  accr[nt]);
        accr[nt] = wmma_f32(ai, nbi, accr[nt]);   // re -= ai*bi
        acci[nt] = wmma_f32(ar, bi,  acci[nt]);
        acci[nt] = wmma_f32(ai, br,  acci[nt]);
      }
    }
  }

  float* Cr = Cre + (size_t)b * NN;
  float* Ci = Cim + (size_t)b * NN;
#pragma unroll
  for (int nt = 0; nt < 4; ++nt) {
#pragma unroll
    for (int j = 0; j < 8; ++j) {   // C/D layout: M = j + half*8, N = ln
      int gm = m0 + wave * 16 + j + half * 8;
      int gn = n0 + nt * 16 + ln;
      size_t off = (size_t)gm * N + gn;
      float re = accr[nt][j];
      float im = acci[nt][j];
      if (EPI == 0) {
        Cr[off] = re; Ci[off] = im;
      } else if (EPI == 1) {
        float mf = (float)mask[(size_t)b * NN + off];
        Cr[off] = re * mf; Ci[off] = im * mf;
      } else {
        size_t co = ((((size_t)b * CC + coil) * NN) + off) * 2;
        float cr = csm[co], cm = csm[co + 1];
        Cr[off] += cr * re + cm * im;   // conj(csm) * img, accumulate
        Ci[off] += cr * im - cm * re;
      }
    }
  }
}

// ------------------------- setup / elementwise -----------------------------

__global__ void fill_F(float* Fre, float* Fim) {
  int idx = blockIdx.x * blockDim.x + threadIdx.x;
  if (idx >= NN) return;
  int j = idx / N, k = idx % N;
  int m = (j * k) % N;                        // DFT phase is periodic in N
  float ang = -2.0f * 3.14159265358979323846f * (float)m / (float)N;
  float inv = 1.0f / sqrtf((float)N);         // ortho norm
  Fre[idx] = cosf(ang) * inv;
  Fim[idx] = sinf(ang) * inv;
}

__global__ void init_state(const float* __restrict__ rhs,
                           float* xr, float* xi, float* rr, float* ri,
                           float* pr, float* pi, float* scal) {
  int idx = blockIdx.x * blockDim.x + threadIdx.x;
  if (idx >= BB * NN) return;
  int b = idx / NN, hw = idx % NN;
  float re = rhs[(size_t)b * 2 * NN + hw];        // rhs[b,0,h,w]
  float im = rhs[(size_t)b * 2 * NN + NN + hw];   // rhs[b,1,h,w]
  xr[idx] = 0.f; xi[idx] = 0.f;
  rr[idx] = re;  ri[idx] = im;
  pr[idx] = re;  pi[idx] = im;
  if (idx == 0) scal[40] = 0.0f;   // done flag
}

__global__ void init_Ap(float* apr, float* api,
                        const float* pr, const float* pi,
                        const float* __restrict__ lam, const float* done) {
  if (done[0] != 0.f) return;
  int idx = blockIdx.x * blockDim.x + threadIdx.x;
  if (idx >= BB * NN) return;
  float l = lam[0];
  apr[idx] = l * pr[idx];
  api[idx] = l * pi[idx];
}

__global__ void coilmul(float* tr, float* ti,
                        const float* pr, const float* pi,
                        const float* __restrict__ csm, int coil, const float* done) {
  if (done[0] != 0.f) return;
  int idx = blockIdx.x * blockDim.x + threadIdx.x;
  if (idx >= BB * NN) return;
  int b = idx / NN, hw = idx % NN;
  size_t co = ((((size_t)b * CC + coil) * NN) + hw) * 2;
  float cr = csm[co], cm = csm[co + 1];
  float ar = pr[idx], ai = pi[idx];
  tr[idx] = cr * ar - cm * ai;
  ti[idx] = cr * ai + cm * ar;
}

__global__ void axpy1(float* xr, float* xi, float* rr, float* ri,
                      const float* pr, const float* pi,
                      const float* apr, const float* api, const float* scal) {
  if (scal[40] != 0.f) return;
  int idx = blockIdx.x * blockDim.x + threadIdx.x;
  if (idx >= BB * NN) return;
  int b = idx / NN;
  float a = scal[24 + b];           // alpha[b]
  xr[idx] += a * pr[idx];  xi[idx] += a * pi[idx];
  rr[idx] -= a * apr[idx]; ri[idx] -= a * api[idx];
}

__global__ void axpy2(float* pr, float* pi,
                      const float* rr, const float* ri, const float* scal) {
  if (scal[40] != 0.f) return;
  int idx = blockIdx.x * blockDim.x + threadIdx.x;
  if (idx >= BB * NN) return;
  int b = idx / NN;
  float bt = scal[32 + b];          // beta[b]
  pr[idx] = rr[idx] + bt * pr[idx];
  pi[idx] = ri[idx] + bt * pi[idx];
}

__global__ void write_out(float* __restrict__ out,
                          const float* xr, const float* xi) {
  int idx = blockIdx.x * blockDim.x + threadIdx.x;
  if (idx >= BB * NN) return;
  out[(size_t)idx * 2 + 0] = xr[idx];   // [B,H,W,2]
  out[(size_t)idx * 2 + 1] = xi[idx];
}

// ------------------------- reductions / scalars ----------------------------

__global__ void __launch_bounds__(512)
dot_stage1(const float* ar, const float* ai, const float* br, const float* bi,
           float* partial, const float* done) {
  if (done[0] != 0.f) return;
  __shared__ float s[512];
  int b = blockIdx.y;
  int i = blockIdx.x * 512 + threadIdx.x;      // 200*512 == NN exactly
  size_t idx = (size_t)b * NN + i;
  s[threadIdx.x] = ar[idx] * br[idx] + ai[idx] * bi[idx];
  __syncthreads();
  for (int st = 256; st > 0; st >>= 1) {
    if (threadIdx.x < st) s[threadIdx.x] += s[threadIdx.x + st];
    __syncthreads();
  }
  if (threadIdx.x == 0) partial[b * 200 + blockIdx.x] = s[0];
}

__global__ void __launch_bounds__(256)
dot_stage2(const float* partial, float* out, const float* done) {
  if (done[0] != 0.f) return;
  __shared__ float s[256];
  int b = blockIdx.x;
  float v = 0.f;
  for (int i = threadIdx.x; i < 200; i += 256) v += partial[b * 200 + i];
  s[threadIdx.x] = v;
  __syncthreads();
  for (int st = 128; st > 0; st >>= 1) {
    if (threadIdx.x < st) s[threadIdx.x] += s[threadIdx.x + st];
    __syncthreads();
  }
  if (threadIdx.x == 0) out[b] = s[0];
}

// scal layout: [0..7]=rTr  [8..15]=pAp  [16..23]=rTr_new  [24..31]=alpha
//              [32..39]=beta  [40]=done
__global__ void scalar_alpha(float* scal) {
  if (scal[40] != 0.f) return;
  for (int b = 0; b < BB; ++b) scal[24 + b] = scal[b] / scal[8 + b];
}
__global__ void scalar_beta(float* scal) {
  if (scal[40] != 0.f) return;
  for (int b = 0; b < BB; ++b) { scal[32 + b] = scal[16 + b] / scal[b]; scal[b] = scal[16 + b]; }
}
__global__ void scalar_done(float* scal) {
  if (scal[40] != 0.f) return;
  float m = 0.f;
  for (int b = 0; b < BB; ++b) m = fmaxf(m, scal[16 + b]);
  if (m < EPSV) scal[40] = 1.f;
}

// ------------------------------- driver ------------------------------------

extern "C" void kernel_launch(void* const* d_in, const int* in_sizes, int n_in,
                              void* d_out, int out_size, void* d_ws, size_t ws_size,
                              hipStream_t stream) {
  (void)in_sizes; (void)n_in; (void)out_size; (void)ws_size;
  const float* rhs  = (const float*)d_in[0];   // [B,2,H,W]
  const float* csm  = (const float*)d_in[1];   // [B,C,H,W,2]
  const int*   mask = (const int*)d_in[2];     // [B,H,W]
  const float* lam  = (const float*)d_in[3];   // [1]

  float* w = (float*)d_ws;
  size_t o = 0;
  float* Fre = w + o; o += NN;       float* Fim = w + o; o += NN;
  float* xr  = w + o; o += BB * NN;  float* xi  = w + o; o += BB * NN;
  float* rr_ = w + o; o += BB * NN;  float* ri  = w + o; o += BB * NN;
  float* pr  = w + o; o += BB * NN;  float* pi  = w + o; o += BB * NN;
  float* apr = w + o; o += BB * NN;  float* api = w + o; o += BB * NN;
  float* tr  = w + o; o += BB * NN;  float* ti  = w + o; o += BB * NN;
  float* ur  = w + o; o += BB * NN;  float* ui  = w + o; o += BB * NN;
  float* partial = w + o; o += BB * 256;
  float* scal    = w + o; o += 64;
  float* done = scal + 40;

  const int EWB = (BB * NN + 255) / 256;
  fill_F<<<(NN + 255) / 256, 256, 0, stream>>>(Fre, Fim);
  init_state<<<EWB, 256, 0, stream>>>(rhs, xr, xi, rr_, ri, pr, pi, scal);

  dim3 rgrid(200, BB);
  dot_stage1<<<rgrid, 512, 0, stream>>>(rr_, ri, rr_, ri, partial, done);
  dot_stage2<<<BB, 256, 0, stream>>>(partial, scal + 0, done);   // rTr0

  dim3 ggrid(5, 5, BB);   // 64x64 tiles over 320x320, batched over B
  for (int it = 0; it < NITER; ++it) {
    init_Ap<<<EWB, 256, 0, stream>>>(apr, api, pr, pi, lam, done);
    for (int c = 0; c < CC; ++c) {
      coilmul<<<EWB, 256, 0, stream>>>(tr, ti, pr, pi, csm, c, done);
      // U = F * T
      zgemm_wmma<0,0,0><<<ggrid, 128, 0, stream>>>(Fre, Fim, 0u, tr, ti, (unsigned)NN,
                                                   ur, ui, mask, csm, c, done);
      // K = mask .* (U * F)
      zgemm_wmma<0,0,1><<<ggrid, 128, 0, stream>>>(ur, ui, (unsigned)NN, Fre, Fim, 0u,
                                                   tr, ti, mask, csm, c, done);
      // V = conj(F) * K
      zgemm_wmma<1,0,0><<<ggrid, 128, 0, stream>>>(Fre, Fim, 0u, tr, ti, (unsigned)NN,
                                                   ur, ui, mask, csm, c, done);
      // Ap += conj(csm) .* (V * conj(F))
      zgemm_wmma<0,1,2><<<ggrid, 128, 0, stream>>>(ur, ui, (unsigned)NN, Fre, Fim, 0u,
                                                   apr, api, mask, csm, c, done);
    }
    dot_stage1<<<rgrid, 512, 0, stream>>>(pr, pi, apr, api, partial, done);
    dot_stage2<<<BB, 256, 0, stream>>>(partial, scal + 8, done);   // pAp
    scalar_alpha<<<1, 1, 0, stream>>>(scal);
    axpy1<<<EWB, 256, 0, stream>>>(xr, xi, rr_, ri, pr, pi, apr, api, scal);
    dot_stage1<<<rgrid, 512, 0, stream>>>(rr_, ri, rr_, ri, partial, done);
    dot_stage2<<<BB, 256, 0, stream>>>(partial, scal + 16, done);  // rTr_new
    scalar_beta<<<1, 1, 0, stream>>>(scal);
    axpy2<<<EWB, 256, 0, stream>>>(pr, pi, rr_, ri, scal);
    scalar_done<<<1, 1, 0, stream>>>(scal);
  }
  write_out<<<EWB, 256, 0, stream>>>((float*)d_out, xr, xi);
}